// ODEFunc_39316130628235
// MI455X (gfx1250) — compile-verified
//
#include <hip/hip_runtime.h>

typedef __attribute__((ext_vector_type(2))) float v2f;
typedef __attribute__((ext_vector_type(8))) float v8f;

#define LN_EPS  1e-5f
#define DAMP    0.1f

// ---------------------------------------------------------------- degrees
__global__ void degree_kernel(const int* __restrict__ dstp,
                              const int* __restrict__ dstn,
                              int* __restrict__ degp,
                              int* __restrict__ degn, int E) {
    int i = blockIdx.x * blockDim.x + threadIdx.x;
    if (i < E) {
        atomicAdd(&degp[dstp[i]], 1);
        atomicAdd(&degn[dstn[i]], 1);
    }
}

__global__ void invsqrt_kernel(const int* __restrict__ degp,
                               const int* __restrict__ degn,
                               float* __restrict__ invp,
                               float* __restrict__ invn, int n) {
    int i = blockIdx.x * blockDim.x + threadIdx.x;
    if (i < n) {
        invp[i] = rsqrtf((float)degp[i] + 1.0f);   // +1 = self loop; deg>=1 always
        invn[i] = rsqrtf((float)degn[i] + 1.0f);
    }
}

// ------------------------------------- fused LayerNorm + hn@W_pos, hn@W_neg
// One wave per 16-row tile. fp32 WMMA 16x16x4, K=32 in 8 steps, 2 N-tiles,
// 2 weight matrices => 32 v_wmma per wave.
__global__ void ln_gemm_kernel(const float* __restrict__ h,
                               const float* __restrict__ gamma,
                               const float* __restrict__ beta,
                               const float* __restrict__ Wp,
                               const float* __restrict__ Wn,
                               float* __restrict__ xwp,
                               float* __restrict__ xwn,
                               int n, int nTiles) {
    const int lane = threadIdx.x & 31;
    const int tile = blockIdx.x * (blockDim.x >> 5) + (threadIdx.x >> 5);
    if (tile >= nTiles) return;                   // wave-uniform exit (EXEC stays full)
    const int hi = lane >> 4;                     // which K-pair half this lane holds
    const int lo = lane & 15;
    const int row0 = tile << 4;
    int row = row0 + lo; if (row >= n) row = n - 1;

    // A-fragment: lane holds cols {4k+2*hi, 4k+2*hi+1} of its row (ISA 16x4 f32 layout)
    const float2* h2 = reinterpret_cast<const float2*>(h);
    v2f a[8];
    float s = 0.f, s2 = 0.f;
#pragma unroll
    for (int k = 0; k < 8; ++k) {
        float2 x = h2[row * 16 + 2 * k + hi];
        a[k].x = x.x; a[k].y = x.y;
        s += x.x + x.y; s2 += x.x * x.x + x.y * x.y;
    }
    // row stats: lane L and L^16 together hold the full 32-col row
    s  += __shfl_xor(s, 16, 32);
    s2 += __shfl_xor(s2, 16, 32);
    const float mu   = s * 0.03125f;
    const float var  = s2 * 0.03125f - mu * mu;
    const float rstd = rsqrtf(var + LN_EPS);

    const float2* g2 = reinterpret_cast<const float2*>(gamma);
    const float2* b2 = reinterpret_cast<const float2*>(beta);
#pragma unroll
    for (int k = 0; k < 8; ++k) {
        float2 g = g2[2 * k + hi], b = b2[2 * k + hi];
        a[k].x = (a[k].x - mu) * rstd * g.x + b.x;
        a[k].y = (a[k].y - mu) * rstd * g.y + b.y;
    }

    v8f cp0 = {0,0,0,0,0,0,0,0}, cp1 = cp0, cn0 = cp0, cn1 = cp0;
#pragma unroll
    for (int k = 0; k < 8; ++k) {
        // B-fragment: lanes 0-15 hold K rows 4k,4k+1; lanes 16-31 hold 4k+2,4k+3
        const int r0 = (4 * k + 2 * hi) * 32;
        v2f bp0, bp1, bn0, bn1;
        bp0.x = Wp[r0 + lo];      bp0.y = Wp[r0 + 32 + lo];
        bp1.x = Wp[r0 + 16 + lo]; bp1.y = Wp[r0 + 48 + lo];
        bn0.x = Wn[r0 + lo];      bn0.y = Wn[r0 + 32 + lo];
        bn1.x = Wn[r0 + 16 + lo]; bn1.y = Wn[r0 + 48 + lo];
        cp0 = __builtin_amdgcn_wmma_f32_16x16x4_f32(false, a[k], false, bp0, (short)0, cp0, false, false);
        cp1 = __builtin_amdgcn_wmma_f32_16x16x4_f32(false, a[k], false, bp1, (short)0, cp1, false, false);
        cn0 = __builtin_amdgcn_wmma_f32_16x16x4_f32(false, a[k], false, bn0, (short)0, cn0, false, false);
        cn1 = __builtin_amdgcn_wmma_f32_16x16x4_f32(false, a[k], false, bn1, (short)0, cn1, false, false);
    }
    // D layout: VGPR r -> row r (lanes 0-15) / row r+8 (lanes 16-31), col = lo
#pragma unroll
    for (int r = 0; r < 8; ++r) {
        int orow = row0 + r + 8 * hi;
        if (orow < n) {
            xwp[orow * 32 + lo]      = cp0[r];
            xwp[orow * 32 + 16 + lo] = cp1[r];
            xwn[orow * 32 + lo]      = cn0[r];
            xwn[orow * 32 + 16 + lo] = cn1[r];
        }
    }
}

// ---------------------------------------------------------------- scatter
// One wave32 per edge; lane = feature. agg[dst] += xw[src] * inv[src]*inv[dst]
// Native no-return fp32 atomic at device scope (RMW resolves in L2), forced
// via inline CDNA5 asm so we never fall back to a CAS loop. GVS addressing:
// saddr = uniform agg base, vaddr = per-lane 32-bit byte offset.
__global__ void scatter_kernel(const int* __restrict__ src,
                               const int* __restrict__ dst,
                               const float* __restrict__ inv,
                               const float* __restrict__ xw,
                               float* __restrict__ agg, int E) {
    int e = (blockIdx.x * blockDim.x + threadIdx.x) >> 5;
    int lane = threadIdx.x & 31;
    if (e >= E) return;
    int sN = src[e], dN = dst[e];
    float coef = inv[sN] * inv[dN];
    float v = xw[sN * 32 + lane] * coef;
    unsigned off = (unsigned)(dN * 32 + lane) * 4u;
    asm volatile("global_atomic_add_f32 %0, %1, %2 scope:SCOPE_DEV"
                 :
                 : "v"(off), "v"(v), "s"(agg)
                 : "memory");
    // s_endpgm performs an implicit wait-idle, covering outstanding STOREcnt.
}

// ------------------- final: self-loop + bias, psi GEMM, tanh, damp, clip
__global__ void final_kernel(const float* __restrict__ h,
                             const float* __restrict__ gamma,
                             const float* __restrict__ beta,
                             const float* __restrict__ invp,
                             const float* __restrict__ invn,
                             const float* __restrict__ xwp,
                             const float* __restrict__ xwn,
                             const float* __restrict__ aggp,
                             const float* __restrict__ aggn,
                             const float* __restrict__ bp,
                             const float* __restrict__ bn,
                             const float* __restrict__ Wpsi,
                             float* __restrict__ out, int n) {
    __shared__ float sW[64 * 32];
    __shared__ float sH[8][64];
    for (int i = threadIdx.x; i < 64 * 32; i += blockDim.x) sW[i] = Wpsi[i];

    const int lane = threadIdx.x & 31;
    const int wv   = threadIdx.x >> 5;
    const int node = blockIdx.x * 8 + wv;
    const bool active = node < n;

    float hp = 0.f, hq = 0.f, x = 0.f;
    if (active) {
        float ivp = invp[node], ivn = invn[node];
        int base = node * 32 + lane;
        hp = aggp[base] + xwp[base] * ivp * ivp + bp[lane];
        hq = aggn[base] + xwn[base] * ivn * ivn + bn[lane];
        x  = h[base];
    }
    sH[wv][lane]      = hp;
    sH[wv][32 + lane] = hq;

    // recompute LayerNorm of h (cheaper than storing hn)
    float s = x, s2 = x * x;
#pragma unroll
    for (int off = 16; off > 0; off >>= 1) {
        s  += __shfl_xor(s, off, 32);
        s2 += __shfl_xor(s2, off, 32);
    }
    float mu  = s * 0.03125f;
    float var = s2 * 0.03125f - mu * mu;
    float hnv = (x - mu) * rsqrtf(var + LN_EPS) * gamma[lane] + beta[lane];

    __syncthreads();

    float acc = 0.f;
#pragma unroll 8
    for (int c = 0; c < 64; ++c) acc += sH[wv][c] * sW[c * 32 + lane];

    float dh = tanhf(acc) - DAMP * hnv;
    dh = fminf(fmaxf(dh, -50.f), 50.f);
    if (active) out[node * 32 + lane] = dh;
}

// ----------------------------------------------------------------- launch
extern "C" void kernel_launch(void* const* d_in, const int* in_sizes, int n_in,
                              void* d_out, int out_size, void* d_ws, size_t ws_size,
                              hipStream_t stream) {
    const float* h     = (const float*)d_in[1];
    const int*   eip   = (const int*)d_in[2];
    const int*   ein   = (const int*)d_in[3];
    const float* gamma = (const float*)d_in[4];
    const float* beta  = (const float*)d_in[5];
    const float* W_pos = (const float*)d_in[6];
    const float* b_pos = (const float*)d_in[7];
    const float* W_neg = (const float*)d_in[8];
    const float* b_neg = (const float*)d_in[9];
    const float* W_psi = (const float*)d_in[10];

    const int N = in_sizes[1] / 32;
    const int E = in_sizes[2] / 2;
    const int* src_p = eip;     const int* dst_p = eip + E;
    const int* src_n = ein;     const int* dst_n = ein + E;

    float* ws      = (float*)d_ws;
    float* agg_pos = ws;                               // 32N
    float* agg_neg = agg_pos + (size_t)32 * N;         // 32N
    float* xw_pos  = agg_neg + (size_t)32 * N;         // 32N
    float* xw_neg  = xw_pos  + (size_t)32 * N;         // 32N
    int*   deg_pos = (int*)(xw_neg + (size_t)32 * N);  // N
    int*   deg_neg = deg_pos + N;                      // N
    float* inv_pos = (float*)(deg_neg + N);            // N
    float* inv_neg = inv_pos + N;                      // N

    hipMemsetAsync(agg_pos, 0, (size_t)64 * N * sizeof(float), stream);
    hipMemsetAsync(deg_pos, 0, (size_t)2 * N * sizeof(int), stream);

    degree_kernel<<<(E + 255) / 256, 256, 0, stream>>>(dst_p, dst_n, deg_pos, deg_neg, E);
    invsqrt_kernel<<<(N + 255) / 256, 256, 0, stream>>>(deg_pos, deg_neg, inv_pos, inv_neg, N);

    const int nTiles = (N + 15) / 16;
    ln_gemm_kernel<<<(nTiles + 7) / 8, 256, 0, stream>>>(h, gamma, beta, W_pos, W_neg,
                                                         xw_pos, xw_neg, N, nTiles);

    long long tot = (long long)E * 32;
    int sblocks = (int)((tot + 255) / 256);
    scatter_kernel<<<sblocks, 256, 0, stream>>>(src_p, dst_p, inv_pos, xw_pos, agg_pos, E);
    scatter_kernel<<<sblocks, 256, 0, stream>>>(src_n, dst_n, inv_neg, xw_neg, agg_neg, E);

    final_kernel<<<(N + 7) / 8, 256, 0, stream>>>(h, gamma, beta, inv_pos, inv_neg,
                                                  xw_pos, xw_neg, agg_pos, agg_neg,
                                                  b_pos, b_neg, W_psi, (float*)d_out, N);
}